// PointCloudTransformerLayer_32547262169514
// MI455X (gfx1250) — compile-verified
//
#include <hip/hip_runtime.h>

typedef __attribute__((ext_vector_type(2))) float v2f;
typedef __attribute__((ext_vector_type(8))) float v8f;

#define TM 16
#define KOFF_PAD 28  // 27 offsets + one always-miss pad column
#define WAVES_PER_BLOCK 4
#define BLOCK_THREADS (WAVES_PER_BLOCK * 32)

// ---------------------------------------------------------------------------
// Per-level voxel keys: key = (x*D + y)*D + z.  Max value 512^3-1 < 2^31, so
// int32 keys are exact and halve search bandwidth vs the reference's int64.
// coords come from sorted np.unique keys -> keys array is sorted ascending.
// ---------------------------------------------------------------------------
__global__ void compute_keys_kernel(const int* __restrict__ coords, int n, int D,
                                    int* __restrict__ keys) {
  int i = blockIdx.x * blockDim.x + threadIdx.x;
  if (i < n) {
    int x = coords[3 * i + 0];
    int y = coords[3 * i + 1];
    int z = coords[3 * i + 2];
    keys[i] = (x * D + y) * D + z;
  }
}

// ---------------------------------------------------------------------------
// One level of the sparse conv as a single gather-GEMM:
//   out[Nout, Cout] = A[Nout, 27*Cin] @ Wflat[27*Cin, Cout]
// Phase 1: 16x28 searchsorted index table -> LDS (col 27 = always miss).
// Phase 2: per-wave 16x16 output tile via exact-fp32 V_WMMA_F32_16X16X4_F32.
// Per offset: stage ALL A/B fragments into registers (clustered loads, one
// wait), then an uninterrupted back-to-back WMMA chain.
// ---------------------------------------------------------------------------
template <int CINL2>
__global__ __launch_bounds__(BLOCK_THREADS)
void sparse_conv_wmma_kernel(const float* __restrict__ Fin,
                             const float* __restrict__ Wf,  // [27*Cin, Cout]
                             const int* __restrict__ coords_out,
                             const int* __restrict__ keys_in,
                             float* __restrict__ out,
                             const float* __restrict__ pos_emb,  // null except last level
                             int n_in, int n_out, int Cout, int D_in) {
  constexpr int Cin = 1 << CINL2;
  __shared__ int s_idx[TM * KOFF_PAD];

  const int m0 = blockIdx.x * TM;
  const int tid = (int)threadIdx.x;

  // ---- Phase 1: searchsorted hash lookup for 16 points x 27 offsets ----
  for (int e = tid; e < TM * KOFF_PAD; e += BLOCK_THREADS) {
    int ii = e / KOFF_PAD;
    int k = e - ii * KOFF_PAD;
    int id = -1;  // pad column k==27 stays -1
    int i = m0 + ii;
    if (i < n_out && k < 27) {
      int dx = k / 9 - 1;
      int dy = (k / 3) % 3 - 1;
      int dz = k % 3 - 1;
      int qx = coords_out[3 * i + 0] * 2 + dx;
      int qy = coords_out[3 * i + 1] * 2 + dy;
      int qz = coords_out[3 * i + 2] * 2 + dz;
      if ((unsigned)qx < (unsigned)D_in && (unsigned)qy < (unsigned)D_in &&
          (unsigned)qz < (unsigned)D_in) {
        int qk = (qx * D_in + qy) * D_in + qz;
        int lo = 0, hi = n_in;  // lower_bound == searchsorted(..., 'left')
        while (lo < hi) {
          int mid = (lo + hi) >> 1;
          if (keys_in[mid] < qk) lo = mid + 1; else hi = mid;
        }
        int p = (lo < n_in) ? lo : (n_in - 1);
        if (keys_in[p] == qk) id = p;
      }
    }
    s_idx[e] = id;
  }
  __syncthreads();

  // ---- Phase 2: WMMA f32 16x16x4 ----
  const int wave = tid >> 5;  // wave32
  const int lane = tid & 31;
  const int col0 = (blockIdx.y * WAVES_PER_BLOCK + wave) * 16;
  if (col0 >= Cout) return;  // wave-uniform exit (after the only barrier)

  const int mrow = lane & 15;          // A's M index / B's N index
  const int khalf = (lane >> 4) << 1;  // lanes 0-15: K+{0,1}; lanes 16-31: K+{2,3}
  const int col = col0 + mrow;

  v8f acc = {};  // 16x16 f32 accumulator (8 VGPRs)

  if constexpr (CINL2 == 0) {
    // Level 1: Cin == 1, K == 27 padded to 28; one offset per K slot.
    // Stage all 7 fragment pairs (branchless), then 7 back-to-back WMMAs.
    v2f af[7], bf[7];
#pragma unroll
    for (int s = 0; s < 7; ++s) {
#pragma unroll
      for (int j = 0; j < 2; ++j) {
        const int koff = 4 * s + khalf + j;
        const int id = s_idx[mrow * KOFF_PAD + koff];
        const int idc = (id >= 0) ? id : 0;
        const float av = Fin[idc];
        const int kc = (koff < 27) ? koff : 26;
        const float bv = Wf[(size_t)kc * Cout + col];
        af[s][j] = (id >= 0) ? av : 0.0f;
        bf[s][j] = (koff < 27) ? bv : 0.0f;
      }
    }
#pragma unroll
    for (int s = 0; s < 7; ++s) {
      acc = __builtin_amdgcn_wmma_f32_16x16x4_f32(false, af[s], false, bf[s],
                                                  (short)0, acc, false, false);
    }
  } else {
    // Levels 2-5: gather index + row bases hoist over the whole cin sweep.
    constexpr int NSTEP = Cin / 4;
    for (int koff = 0; koff < 27; ++koff) {
      const int id = s_idx[mrow * KOFF_PAD + koff];
      const int idc = (id >= 0) ? id : 0;  // always-valid row
      const float* arow = Fin + (size_t)idc * Cin + khalf;
      const float* bcol = Wf + (size_t)(koff * Cin + khalf) * Cout + col;
      // next offset's B panel -> global_prefetch_b8 (clamped, unconditional)
      const int kn = (koff + 1 < 27) ? koff + 1 : 26;
      __builtin_prefetch(Wf + (size_t)(kn * Cin) * Cout + col0, 0, 1);

      // Stage: clustered loads, single wait.
      v2f af[NSTEP], bf[NSTEP];
#pragma unroll
      for (int s = 0; s < NSTEP; ++s) {
        af[s] = *(const v2f*)(arow + 4 * s);           // b64 load, contiguous pair
        bf[s][0] = bcol[(size_t)(4 * s) * Cout];       // stride-Cout pair
        bf[s][1] = bcol[(size_t)(4 * s + 1) * Cout];
      }
      // Math: uninterrupted WMMA chain (miss-select folds to cndmask).
#pragma unroll
      for (int s = 0; s < NSTEP; ++s) {
        v2f a;
        a[0] = (id >= 0) ? af[s][0] : 0.0f;
        a[1] = (id >= 0) ? af[s][1] : 0.0f;
        acc = __builtin_amdgcn_wmma_f32_16x16x4_f32(false, a, false, bf[s],
                                                    (short)0, acc, false, false);
      }
    }
  }

  // ---- Store (C/D layout: VGPR r -> rows r (lanes 0-15) / r+8 (lanes 16-31))
#pragma unroll
  for (int r = 0; r < 8; ++r) {
    int row = r + ((lane >> 4) << 3);
    int i = m0 + row;
    if (i < n_out) {
      float v = acc[r];
      if (pos_emb) v += pos_emb[(size_t)i * Cout + col];
      out[(size_t)i * Cout + col] = v;
    }
  }
}

// ---------------------------------------------------------------------------
extern "C" void kernel_launch(void* const* d_in, const int* in_sizes, int n_in,
                              void* d_out, int out_size, void* d_ws, size_t ws_size,
                              hipStream_t stream) {
  (void)n_in; (void)out_size; (void)ws_size;

  const float* features = (const float*)d_in[0];
  const float* W[5];
  for (int i = 0; i < 5; ++i) W[i] = (const float*)d_in[1 + i];
  const float* pos_emb = (const float*)d_in[6];
  const int* coords[6];
  int N[6];
  for (int i = 0; i < 6; ++i) {
    coords[i] = (const int*)d_in[7 + i];
    N[i] = in_sizes[7 + i] / 3;
  }

  static const int Dlev[5] = {512, 256, 128, 64, 32};
  static const int chans[6] = {1, 16, 32, 64, 128, 512};

  // Workspace carve-out: int32 keys per input level + fp32 intermediate features.
  char* ws = (char*)d_ws;
  size_t off = 0;
  auto walloc = [&](size_t bytes) -> void* {
    void* p = (void*)(ws + off);
    off = (off + bytes + 255) & ~(size_t)255;
    return p;
  };
  int* keys[5];
  for (int l = 0; l < 5; ++l) keys[l] = (int*)walloc((size_t)N[l] * sizeof(int));
  float* fbuf[4];
  for (int l = 0; l < 4; ++l)
    fbuf[l] = (float*)walloc((size_t)N[l + 1] * chans[l + 1] * sizeof(float));

  // Keys for all input levels (stream-ordered before their consumers).
  for (int l = 0; l < 5; ++l) {
    int blocks = (N[l] + 255) / 256;
    compute_keys_kernel<<<blocks, 256, 0, stream>>>(coords[l], N[l], Dlev[l], keys[l]);
  }

  // Five gather-GEMM levels; last one writes d_out and fuses the pos_emb add.
  const float* fin = features;
  for (int l = 0; l < 5; ++l) {
    int Cout = chans[l + 1];
    int nout = N[l + 1];
    float* fout = (l == 4) ? (float*)d_out : fbuf[l];
    const float* pe = (l == 4) ? pos_emb : nullptr;
    dim3 grid((nout + TM - 1) / TM,
              (Cout / 16 + WAVES_PER_BLOCK - 1) / WAVES_PER_BLOCK);
    switch (l) {
      case 0:
        sparse_conv_wmma_kernel<0><<<grid, BLOCK_THREADS, 0, stream>>>(
            fin, W[l], coords[l + 1], keys[l], fout, pe, N[l], nout, Cout, Dlev[l]);
        break;
      case 1:
        sparse_conv_wmma_kernel<4><<<grid, BLOCK_THREADS, 0, stream>>>(
            fin, W[l], coords[l + 1], keys[l], fout, pe, N[l], nout, Cout, Dlev[l]);
        break;
      case 2:
        sparse_conv_wmma_kernel<5><<<grid, BLOCK_THREADS, 0, stream>>>(
            fin, W[l], coords[l + 1], keys[l], fout, pe, N[l], nout, Cout, Dlev[l]);
        break;
      case 3:
        sparse_conv_wmma_kernel<6><<<grid, BLOCK_THREADS, 0, stream>>>(
            fin, W[l], coords[l + 1], keys[l], fout, pe, N[l], nout, Cout, Dlev[l]);
        break;
      default:
        sparse_conv_wmma_kernel<7><<<grid, BLOCK_THREADS, 0, stream>>>(
            fin, W[l], coords[l + 1], keys[l], fout, pe, N[l], nout, Cout, Dlev[l]);
        break;
    }
    fin = fout;
  }
}